// CBAMmax_45896020525453
// MI455X (gfx1250) — compile-verified
//
#include <hip/hip_runtime.h>
#include <math.h>

typedef float v2f __attribute__((ext_vector_type(2)));
typedef float v8f __attribute__((ext_vector_type(8)));

#define BB    8
#define CC    64
#define HH    256
#define WW    256
#define HWP   65536      // H*W
#define HP    246
#define WP    246
#define HPWP  60516      // HP*WP
#define NTIL  256        // 16x16 tiles per image
#define LSTR  34         // LDS strip stride (bank-conflict free, 8B-align ok)
#define C1SS  0.01f      // (0.01*10)^2
#define C2SS  0.09f      // (0.03*10)^2

__constant__ float GW[11] = {
    0.00102838f, 0.00759884f, 0.03600077f, 0.10936070f, 0.21300553f,
    0.26601172f, 0.21300553f, 0.10936070f, 0.03600077f, 0.00759884f,
    0.00102838f };

__device__ __forceinline__ float gband(int d) {
    return (d >= 0 && d < 11) ? GW[d] : 0.0f;
}

// D = A(16x4 f32) x B(4x16 f32) + C(16x16 f32)
__device__ __forceinline__ v8f wmma4(v2f a, v2f b, v8f c) {
    return __builtin_amdgcn_wmma_f32_16x16x4_f32(false, a, false, b,
                                                 (short)0, c, false, false);
}

// Per-lane constant Gaussian-band fragments: serve as vertical-pass A-matrix
// AND horizontal-pass B-matrix tiles (same layout by symmetry of G / G^T).
__device__ __forceinline__ void make_gfrags(int lane, v2f gfr[7]) {
    int half = lane >> 4, l15 = lane & 15;
#pragma unroll
    for (int q = 0; q < 7; ++q) {
        int d = 4 * q + 2 * half - l15;
        gfr[q].x = gband(d);
        gfr[q].y = gband(d + 1);
    }
}

// Load B-fragments of a 28x32 input region at (m0, n0) of a 256x256 image.
// Clamped reads only ever feed invalid (unstored) outputs.
__device__ __forceinline__ void load_frags(const float* __restrict__ img,
                                           int m0, int n0, int lane,
                                           v2f xfr[2][7]) {
    int half = lane >> 4, l15 = lane & 15;
#pragma unroll
    for (int t = 0; t < 2; ++t) {
        int c = n0 + 16 * t + l15;  if (c > WW - 1) c = WW - 1;
#pragma unroll
        for (int q = 0; q < 7; ++q) {
            int r  = m0 + 4 * q + 2 * half;
            int r0 = r     > HH - 1 ? HH - 1 : r;
            int r1 = r + 1 > HH - 1 ? HH - 1 : r + 1;
            xfr[t][q].x = img[r0 * WW + c];
            xfr[t][q].y = img[r1 * WW + c];
        }
    }
}

// Full 2D 11-tap blur of one 16x16 output tile via banded GEMM: Out = G X G^T.
// 14 vertical WMMAs (two 16-col Tv tiles) + LDS relayout + 7 horizontal WMMAs.
__device__ __forceinline__ v8f blur_tile(const v2f gfr[7], const v2f xfr[2][7],
                                         float* __restrict__ strip, int lane) {
    int half = lane >> 4, l15 = lane & 15;
#pragma unroll
    for (int t = 0; t < 2; ++t) {
        v8f acc = {};
#pragma unroll
        for (int q = 0; q < 7; ++q) acc = wmma4(gfr[q], xfr[t][q], acc);
        // D layout -> LDS strip: row = r + 8*half, col = l15 + 16*t
#pragma unroll
        for (int r = 0; r < 8; ++r)
            strip[(r + 8 * half) * LSTR + l15 + 16 * t] = acc[r];
    }
    // Intra-wave exchange; compiler inserts s_wait_dscnt for the RAW.
    v8f acc = {};
#pragma unroll
    for (int q = 0; q < 7; ++q) {
        int col = 4 * q + 2 * half;
        v2f a;
        a.x = strip[l15 * LSTR + col];
        a.y = strip[l15 * LSTR + col + 1];
        acc = wmma4(a, gfr[q], acc);  // B-frag == same band constants
    }
    return acc;
}

__device__ __forceinline__ float wave_sum(float v) {
#pragma unroll
    for (int off = 16; off > 0; off >>= 1) v += __shfl_xor(v, off, 32);
    return v;
}

// ---------------- Kernel 1: channel reduce + masked moment sums -------------
__global__ void k_reduce(const float* __restrict__ x, const int* __restrict__ mask,
                         float* __restrict__ xc, float* __restrict__ stats) {
    int t  = blockIdx.x * 256 + threadIdx.x;   // B*H*W threads
    int b  = t >> 16;
    int hw = t & 65535;
    const float* xb = x + ((size_t)b * CC) * HWP + hw;
    float mx = -3.4e38f, mn = 3.4e38f, sm = 0.0f;
#pragma unroll 4
    for (int c = 0; c < CC; ++c) {
        float v = xb[c * HWP];
        mx = fmaxf(mx, v); mn = fminf(mn, v); sm += v;
    }
    float me = sm * (1.0f / CC);
    xc[(b * 3 + 0) * HWP + hw] = mx;
    xc[(b * 3 + 1) * HWP + hw] = me;
    xc[(b * 3 + 2) * HWP + hw] = mn;

    float mv = (float)mask[b * HWP + hw];
    float p[7];
    p[0] = mv;
    p[1] = mx * mv; p[2] = me * mv; p[3] = mn * mv;
    p[4] = mx * mx * mv; p[5] = me * me * mv; p[6] = mn * mn * mv;

    __shared__ float red[8][8];
    int wave = threadIdx.x >> 5, lane = threadIdx.x & 31;
#pragma unroll
    for (int k = 0; k < 7; ++k) {
        float s = wave_sum(p[k]);
        if (lane == 0) red[wave][k] = s;
    }
    __syncthreads();
    if (threadIdx.x < 7) {
        float tot = 0.0f;
#pragma unroll
        for (int w = 0; w < 8; ++w) tot += red[w][threadIdx.x];
        atomicAdd(&stats[(blockIdx.x >> 8) * 8 + threadIdx.x], tot);
    }
}

// ---------------- Kernel 2: finalize mean / inv-std -------------------------
__global__ void k_finalize(const float* __restrict__ stats, float* __restrict__ ms) {
    int t = threadIdx.x;
    if (t < 24) {
        int b = t / 3, j = t % 3;
        float n  = stats[b * 8];
        float s1 = stats[b * 8 + 1 + j];
        float s2 = stats[b * 8 + 4 + j];
        float mean = s1 / n;
        float var  = (s2 - s1 * s1 / n) / (n - 1.0f);
        ms[t * 2]     = mean;
        ms[t * 2 + 1] = rsqrtf(var);
    }
}

// ---------------- Kernel 3: xf = (xc - mean)*invstd*mask --------------------
__global__ void k_xf(const float* __restrict__ xc, const int* __restrict__ mask,
                     const float* __restrict__ ms, float* __restrict__ xf) {
    int t  = blockIdx.x * 256 + threadIdx.x;   // 24*65536 threads
    int im = t >> 16;
    int hw = t & 65535;
    int b  = im / 3;
    float mv   = (float)mask[b * HWP + hw];
    float mean = ms[im * 2], inv = ms[im * 2 + 1];
    xf[t] = (xc[t] - mean) * inv * mv;
}

// ---------------- Kernel 4: mu_f = blur(xf), ef2 = blur(xf^2) ---------------
__global__ void k_blur_small(const float* __restrict__ xf,
                             float* __restrict__ muf, float* __restrict__ ef2) {
    __shared__ float lds[8][16 * LSTR];
    int lane = threadIdx.x & 31, wave = threadIdx.x >> 5;
    int wid  = blockIdx.x * 8 + wave;          // 24*256 waves total (exact)
    int im   = wid >> 8;
    int tile = wid & 255;
    int m0 = (tile >> 4) << 4, n0 = (tile & 15) << 4;
    float* strip = &lds[wave][0];

    v2f gfr[7];            make_gfrags(lane, gfr);
    v2f fr[2][7];          load_frags(xf + (size_t)im * HWP, m0, n0, lane, fr);
    v2f sq[2][7];
#pragma unroll
    for (int t = 0; t < 2; ++t)
#pragma unroll
        for (int q = 0; q < 7; ++q) sq[t][q] = fr[t][q] * fr[t][q];

    v8f bmu = blur_tile(gfr, fr, strip, lane);
    v8f be2 = blur_tile(gfr, sq, strip, lane);

    int half = lane >> 4, l15 = lane & 15;
#pragma unroll
    for (int r = 0; r < 8; ++r) {
        int row = m0 + r + 8 * half, col = n0 + l15;
        if (row < HP && col < WP) {
            int idx = (im * HP + row) * WP + col;
            muf[idx] = bmu[r];
            ef2[idx] = be2[r];
        }
    }
}

// ------- Kernel 5: fused blur(x), blur(x^2), blur(xf_j*x) + SSIM reduce -----
__global__ void k_fused(const float* __restrict__ x, const float* __restrict__ xf,
                        const float* __restrict__ muf, const float* __restrict__ ef2,
                        float* __restrict__ ssum) {
    __shared__ float lds[8][16 * LSTR];
    int lane = threadIdx.x & 31, wave = threadIdx.x >> 5;
    int wid  = blockIdx.x * 8 + wave;          // 512*256 waves total (exact)
    int pair = wid >> 8;                       // (b,c)
    int tile = wid & 255;
    int b = pair >> 6, c = pair & 63;
    int m0 = (tile >> 4) << 4, n0 = (tile & 15) << 4;
    int half = lane >> 4, l15 = lane & 15;
    float* strip = &lds[wave][0];

    const float* ximg = x + (size_t)pair * HWP;
    __builtin_prefetch(ximg + m0 * WW + n0, 0, 1);

    v2f gfr[7];      make_gfrags(lane, gfr);
    v2f xr[2][7];    load_frags(ximg, m0, n0, lane, xr);
    v2f x2[2][7];
#pragma unroll
    for (int t = 0; t < 2; ++t)
#pragma unroll
        for (int q = 0; q < 7; ++q) x2[t][q] = xr[t][q] * xr[t][q];

    v8f mu2t = blur_tile(gfr, xr, strip, lane);   // blur(x)
    v8f ex2t = blur_tile(gfr, x2, strip, lane);   // blur(x^2)

    for (int j = 0; j < 3; ++j) {
        int im3 = b * 3 + j;
        v2f fr[2][7];  load_frags(xf + (size_t)im3 * HWP, m0, n0, lane, fr);
        v2f pr[2][7];
#pragma unroll
        for (int t = 0; t < 2; ++t)
#pragma unroll
            for (int q = 0; q < 7; ++q) pr[t][q] = fr[t][q] * xr[t][q];
        v8f exft = blur_tile(gfr, pr, strip, lane);  // blur(xf_j * x_c)

        float s = 0.0f;
#pragma unroll
        for (int r = 0; r < 8; ++r) {
            int row = m0 + r + 8 * half, col = n0 + l15;
            bool ok = (row < HP) && (col < WP);
            int ridx = ok ? (im3 * HP + row) * WP + col : 0;
            float mu1v = muf[ridx];
            float e2v  = ef2[ridx];
            float mu2v = mu2t[r];
            float s1 = e2v - mu1v * mu1v;
            float s2 = ex2t[r] - mu2v * mu2v;
            float s12 = exft[r] - mu1v * mu2v;
            float num = (2.0f * mu1v * mu2v + C1SS) * (2.0f * s12 + C2SS);
            float den = (mu1v * mu1v + mu2v * mu2v + C1SS) * (s1 + s2 + C2SS);
            float v = num / den;
            s += ok ? v : 0.0f;
        }
        s = wave_sum(s);
        if (lane == 0) atomicAdd(&ssum[(b * 3 + j) * CC + c], s);
    }
}

// ---------------- Kernel 6: head (3x1 conv + MLP + sigmoid) -----------------
__global__ void k_head(const float* __restrict__ ssum, const float* __restrict__ cw,
                       const float* __restrict__ w1, const float* __restrict__ b1,
                       const float* __restrict__ w2, const float* __restrict__ b2,
                       float* __restrict__ out) {
    __shared__ float sinfo[1536];
    __shared__ float hbuf[512];
    __shared__ float h1buf[512];
    int tid = threadIdx.x;  // 512 threads
    for (int i = tid; i < 1536; i += 512) {
        float v = ssum[i] * (1.0f / (float)HPWP);
        sinfo[i] = v;
        out[512 + i] = v;   // ssim_info output
    }
    __syncthreads();
    int b = tid >> 6, o = tid & 63;
    float acc = 0.0f;
#pragma unroll
    for (int i = 0; i < 3; ++i)
#pragma unroll
        for (int k = 0; k < 3; ++k) {
            int cc = o + k - 1;
            if (cc >= 0 && cc < 64) acc += sinfo[(b * 3 + i) * 64 + cc] * cw[i * 3 + k];
        }
    hbuf[tid] = fmaxf(acc, 0.0f);
    __syncthreads();
    acc = b1[o];
    for (int c = 0; c < 64; ++c) acc += hbuf[b * 64 + c] * w1[o * 64 + c];
    h1buf[tid] = fmaxf(acc, 0.0f);
    __syncthreads();
    acc = b2[o];
    for (int c = 0; c < 64; ++c) acc += h1buf[b * 64 + c] * w2[o * 64 + c];
    out[tid] = 1.0f / (1.0f + expf(-acc));
}

extern "C" void kernel_launch(void* const* d_in, const int* in_sizes, int n_in,
                              void* d_out, int out_size, void* d_ws, size_t ws_size,
                              hipStream_t stream) {
    const float* x    = (const float*)d_in[0];
    const int*   mask = (const int*)d_in[1];
    const float* cw   = (const float*)d_in[2];
    const float* w1   = (const float*)d_in[3];
    const float* b1   = (const float*)d_in[4];
    const float* w2   = (const float*)d_in[5];
    const float* b2   = (const float*)d_in[6];
    float* out = (float*)d_out;

    float* ws    = (float*)d_ws;
    float* xc    = ws;                      // 24*65536
    float* xf    = xc  + 24 * HWP;          // 24*65536
    float* muf   = xf  + 24 * HWP;          // 24*60516
    float* ef2   = muf + 24 * HPWP;         // 24*60516
    float* stats = ef2 + 24 * HPWP;         // 64
    float* ms    = stats + 64;              // 48
    float* ssum  = ms + 48;                 // 1536

    hipMemsetAsync(stats, 0, (64 + 48 + 1536) * sizeof(float), stream);

    k_reduce    <<<2048, 256, 0, stream>>>(x, mask, xc, stats);
    k_finalize  <<<1, 32, 0, stream>>>(stats, ms);
    k_xf        <<<6144, 256, 0, stream>>>(xc, mask, ms, xf);
    k_blur_small<<<768, 256, 0, stream>>>(xf, muf, ef2);
    k_fused     <<<16384, 256, 0, stream>>>(x, xf, muf, ef2, ssum);
    k_head      <<<1, 512, 0, stream>>>(ssum, cw, w1, b1, w2, b2, out);
}